// GridEmbedding_15118284882663
// MI455X (gfx1250) — compile-verified
//
#include <hip/hip_runtime.h>
#include <stdint.h>
#include <math.h>

// Problem constants (from reference)
#define GRID_X   128
#define NUM_EMB  (128*128*128)     // 2,097,152 rows
#define EMB_DIM  32
#define N_PTS    2097152

// Launch shape: 2048 blocks * 8 waves = 16384 waves; 2,097,152/16,384 = 128 pts/wave
#define TPB      256
#define WPB      (TPB/32)
#define NBLOCKS  2048
#define NWAVES   (NBLOCKS*WPB)
#define PPW      (N_PTS/NWAVES)    // 128, even

// ---------- CDNA5 async global->LDS gather (ASYNCcnt path) ----------
typedef int vi4 __attribute__((vector_size(16)));
typedef __attribute__((address_space(1))) vi4 gvi4;   // global int4
typedef __attribute__((address_space(3))) vi4 lvi4;   // LDS int4

__device__ __forceinline__ void async_b128(const float* g, float* l) {
#if __has_builtin(__builtin_amdgcn_global_load_async_to_lds_b128)
  __builtin_amdgcn_global_load_async_to_lds_b128(
      (gvi4*)(void*)g,
      (lvi4*)(void*)l,
      /*imm offset*/0, /*cpol*/0);
#else
  unsigned loff = (unsigned)(uintptr_t)(__attribute__((address_space(3))) void*)l;
  asm volatile("global_load_async_to_lds_b128 %0, %1, off"
               :: "v"(loff), "v"(g) : "memory");
#endif
}

__device__ __forceinline__ void wait_async_le4() {
#if __has_builtin(__builtin_amdgcn_s_wait_asynccnt)
  __builtin_amdgcn_s_wait_asynccnt(4);
#else
  asm volatile("s_wait_asynccnt 0x4" ::: "memory");
#endif
}

__device__ __forceinline__ void wait_async_le0() {
#if __has_builtin(__builtin_amdgcn_s_wait_asynccnt)
  __builtin_amdgcn_s_wait_asynccnt(0);
#else
  asm volatile("s_wait_asynccnt 0x0" ::: "memory");
#endif
}

__global__ __launch_bounds__(TPB) void grid_emb_kernel(
    const float* __restrict__ xin, const float* __restrict__ emb,
    float* __restrict__ out)
{
  // [wave][double-buffer][point-in-pair][corner-row][channel] : 32 KB/block
  __shared__ float lbuf[WPB][2][2][8][EMB_DIM];

  const int lane = threadIdx.x & 31;
  const int wv   = threadIdx.x >> 5;
  const int wave = blockIdx.x * WPB + wv;

  const int r = lane >> 3;   // corner-row group 0..3 handled by this lane
  const int e = lane & 7;    // 16B chunk within a 128B row

  // Issue 2 async B128 gathers covering all 8 corner rows of point p into dst.
  auto prefetch_one = [&](int p, float* dst, float& w0, float& w1, float& w2) {
    const float px = __builtin_nontemporal_load(xin + 3*p + 0);
    const float py = __builtin_nontemporal_load(xin + 3*p + 1);
    const float pz = __builtin_nontemporal_load(xin + 3*p + 2);
    float fx = px * 128.0f, fy = py * 128.0f, fz = pz * 128.0f;
    float ix = floorf(fx), iy = floorf(fy), iz = floorf(fz);
    w0 = fx - ix; w1 = fy - iy; w2 = fz - iz;
    int b = (int)(ix + iy * 128.0f + iz * 16384.0f);   // exact: < 2^24
    int c000 = (b >= NUM_EMB) ? (NUM_EMB - 1) : b;
    auto cl = [&](int off) { int c = b + off; return (c >= NUM_EMB) ? c000 : c; };
    // row order (matches reference stack): 000,001,010,100 | 011,101,110,111
    int c001 = cl(1),    c010 = cl(128),   c100 = cl(16384);
    int c011 = cl(129),  c101 = cl(16385), c110 = cl(16512), c111 = cl(16513);
    int lo01  = (r & 1) ? c001 : c000;
    int lo23  = (r & 1) ? c100 : c010;
    int ci_lo = (r & 2) ? lo23 : lo01;        // rows 0..3
    int hi01  = (r & 1) ? c101 : c011;
    int hi23  = (r & 1) ? c111 : c110;
    int ci_hi = (r & 2) ? hi23 : hi01;        // rows 4..7
    const float* glo = emb + (size_t)ci_lo * EMB_DIM + (e << 2);
    const float* ghi = emb + (size_t)ci_hi * EMB_DIM + (e << 2);
    async_b128(glo, dst + (r    ) * EMB_DIM + (e << 2));
    async_b128(ghi, dst + (r + 4) * EMB_DIM + (e << 2));
  };

  // Trilinear blend from LDS; lane = channel -> conflict-free ds_load_b32.
  auto blend_one = [&](int p, const float* L, float w0, float w1, float w2) {
    float v000 = L[0*EMB_DIM + lane];
    float v001 = L[1*EMB_DIM + lane];
    float v010 = L[2*EMB_DIM + lane];
    float v100 = L[3*EMB_DIM + lane];
    float v011 = L[4*EMB_DIM + lane];
    float v101 = L[5*EMB_DIM + lane];
    float v110 = L[6*EMB_DIM + lane];
    float v111 = L[7*EMB_DIM + lane];
    float c00 = fmaf(w0, v100 - v000, v000);
    float c01 = fmaf(w0, v101 - v001, v001);
    float c10 = fmaf(w0, v110 - v010, v010);
    float c11 = fmaf(w0, v111 - v011, v011);
    float c0  = fmaf(w1, c10 - c00, c00);
    float c1  = fmaf(w1, c11 - c01, c01);
    float res = fmaf(w2, c1 - c0, c0);
    __builtin_nontemporal_store(res, out + (size_t)p * EMB_DIM + lane);
  };

  int p = wave * PPW;
  float a00=0,a01=0,a02=0, a10=0,a11=0,a12=0;   // weights for current pair
  float b00=0,b01=0,b02=0, b10=0,b11=0,b12=0;   // weights for prefetched pair
  prefetch_one(p + 0, &lbuf[wv][0][0][0][0], a00, a01, a02);
  prefetch_one(p + 1, &lbuf[wv][0][1][0][0], a10, a11, a12);
  int buf = 0;

  #pragma unroll 1
  for (int it = 0; it < PPW/2; ++it) {
    bool more = (it + 1 < PPW/2);
    if (more) {
      prefetch_one(p + 2, &lbuf[wv][buf ^ 1][0][0][0], b00, b01, b02);
      prefetch_one(p + 3, &lbuf[wv][buf ^ 1][1][0][0], b10, b11, b12);
      wait_async_le4();   // newest 4 asyncs are the next pair; current pair done
    } else {
      wait_async_le0();
    }
    asm volatile("" ::: "memory");   // keep LDS reads after the wait
    blend_one(p + 0, &lbuf[wv][buf][0][0][0], a00, a01, a02);
    blend_one(p + 1, &lbuf[wv][buf][1][0][0], a10, a11, a12);
    a00=b00; a01=b01; a02=b02; a10=b10; a11=b11; a12=b12;
    p += 2; buf ^= 1;
  }
}

extern "C" void kernel_launch(void* const* d_in, const int* in_sizes, int n_in,
                              void* d_out, int out_size, void* d_ws, size_t ws_size,
                              hipStream_t stream) {
  const float* x   = (const float*)d_in[0];   // (N_PTS, 3) f32
  const float* emb = (const float*)d_in[1];   // (NUM_EMB, 32) f32
  float* out       = (float*)d_out;           // (N_PTS, 32) f32
  grid_emb_kernel<<<dim3(NBLOCKS), dim3(TPB), 0, stream>>>(x, emb, out);
}